// DidgeModel_30270929502339
// MI455X (gfx1250) — compile-verified
//
#include <hip/hip_runtime.h>
#include <hip/hip_bf16.h>

typedef __attribute__((ext_vector_type(2))) float v2f;
typedef __attribute__((ext_vector_type(8))) float v8f;

#define RHO_AIR 1.2929f
#define C_SND   343.37f
#define N_SUB   128
#define PI_F    3.14159265358979f

// d += a * b  (complex fused multiply-add)
__device__ __forceinline__ void cmad(float& dr, float& di,
                                     float ar, float ai, float br, float bi) {
    dr = fmaf(ar, br, fmaf(-ai, bi, dr));
    di = fmaf(ar, bi, fmaf( ai, br, di));
}

__global__ __launch_bounds__(32)
void didge_impedance_kernel(const float* __restrict__ plen,
                            const float* __restrict__ pd1,
                            const int*   __restrict__ pfmin,
                            float* __restrict__ out, int F)
{
    const int fi   = blockIdx.x;
    if (fi >= F) return;
    const int lane = threadIdx.x;            // 0..31, one wave per frequency

    const float length = plen[0];
    const float d1     = pd1[0];
    const float f      = (float)(pfmin[0] + fi);

    const float d0  = 32.0f;
    const float L_m = length * 10.0f * (1.0f / 1000.0f);
    const float dL  = L_m * (1.0f / (float)N_SUB);
    const float k   = 2.0f * PI_F * f / C_SND;
    const float afc = 3e-5f * sqrtf(f);      // alpha numerator
    const float b   = k * dL;                // imag part of gamma*dL (segment-invariant)
    float sb, cb;
    __sincosf(b, &sb, &cb);

    // ---- lane-private ordered partial product of 4 segment matrices (complex 2x2) ----
    float p00r = 1.f, p00i = 0.f, p01r = 0.f, p01i = 0.f;
    float p10r = 0.f, p10i = 0.f, p11r = 1.f, p11i = 0.f;

#pragma unroll
    for (int q = 0; q < 4; ++q) {
        const int   s  = lane * 4 + q;
        const float t  = ((float)s + 0.5f) * (1.0f / (float)N_SUB);
        const float r  = (d0 + (d1 - d0) * t) * (1.0f / 2000.0f);
        const float a  = (afc / r) * dL;                 // real part of gamma*dL (tiny)
        const float Z0 = RHO_AIR * C_SND / (PI_F * r * r);
        const float iZ0 = 1.0f / Z0;

        // accurate small-argument cosh/sinh via expm1
        const float u   = expm1f(a);
        const float sha = 0.5f * (u + u / (1.0f + u));
        const float cha = 1.0f + 0.5f * u * u / (1.0f + u);

        const float chr = cha * cb, chi = sha * sb;      // cosh(a+ib)
        const float shr = sha * cb, shi = cha * sb;      // sinh(a+ib)

        const float t00r = chr,      t00i = chi;
        const float t01r = Z0 * shr, t01i = Z0 * shi;
        const float t10r = shr * iZ0, t10i = shi * iZ0;
        const float t11r = chr,      t11i = chi;

        // P = P @ T  (ordered, ascending segment index)
        float n00r = 0.f, n00i = 0.f, n01r = 0.f, n01i = 0.f;
        float n10r = 0.f, n10i = 0.f, n11r = 0.f, n11i = 0.f;
        cmad(n00r, n00i, p00r, p00i, t00r, t00i); cmad(n00r, n00i, p01r, p01i, t10r, t10i);
        cmad(n01r, n01i, p00r, p00i, t01r, t01i); cmad(n01r, n01i, p01r, p01i, t11r, t11i);
        cmad(n10r, n10i, p10r, p10i, t00r, t00i); cmad(n10r, n10i, p11r, p11i, t10r, t10i);
        cmad(n11r, n11i, p10r, p10i, t01r, t01i); cmad(n11r, n11i, p11r, p11i, t11r, t11i);
        p00r = n00r; p00i = n00i; p01r = n01r; p01i = n01i;
        p10r = n10r; p10i = n10i; p11r = n11r; p11i = n11i;
    }

    // ---- stage partials into LDS as 4x4 real embeddings of complex 2x2 ----
    // R([[A,B],[C,D]]) rows: [Ar,-Ai,Br,-Bi],[Ai,Ar,Bi,Br],[Cr,-Ci,Dr,-Di],[Ci,Cr,Di,Dr]
    __shared__ float mats[48][17];           // stride 17 -> conflict-free on 64 banks
    {
        float* m = mats[lane];
        m[0]  =  p00r; m[1]  = -p00i; m[2]  =  p01r; m[3]  = -p01i;
        m[4]  =  p00i; m[5]  =  p00r; m[6]  =  p01i; m[7]  =  p01r;
        m[8]  =  p10r; m[9]  = -p10i; m[10] =  p11r; m[11] = -p11i;
        m[12] =  p10i; m[13] =  p10r; m[14] =  p11i; m[15] =  p11r;
    }
    __syncthreads();

    // ---- log2(32)=5-level ordered tree combine via V_WMMA_F32_16X16X4_F32 ----
    // Pack 4 left 4x4s as row-blocks of A (16x4) and 4 right 4x4s as column-blocks
    // of B (4x16); D's diagonal blocks are the 4 independent products (no crosstalk).
    const int mrow = lane & 15;
    const int ablk = mrow >> 2, ar = mrow & 3;     // A: lane = row M
    const int kb   = (lane >> 4) << 1;             // lane half selects K = {0,1} or {2,3}
    const int col  = lane & 15;
    const int bblk = col >> 2, bc = col & 3;       // B: lane = col N
    const bool hi  = lane >= 16;

    int src = 0, dst = 32, n = 32;
    while (n > 1) {
        const int pairs = n >> 1;
        const int nw = (pairs + 3) >> 2;           // 4 pair-products per WMMA
        for (int w = 0; w < nw; ++w) {
            int pa = w * 4 + ablk;                 // pair index feeding A rows
            int lj = 2 * pa;     if (lj > n - 2) lj = n - 2;   // clamp (output discarded)
            int pb = w * 4 + bblk;                 // pair index feeding B cols
            int rj = 2 * pb + 1; if (rj > n - 1) rj = n - 1;

            v2f av, bv;
            av.x = mats[src + lj][ar * 4 + kb];
            av.y = mats[src + lj][ar * 4 + kb + 1];
            bv.x = mats[src + rj][kb * 4 + bc];
            bv.y = mats[src + rj][(kb + 1) * 4 + bc];

            v8f cz = {0.f, 0.f, 0.f, 0.f, 0.f, 0.f, 0.f, 0.f};
            v8f d = __builtin_amdgcn_wmma_f32_16x16x4_f32(
                        false, av, false, bv, (short)0, cz, false, false);

            // write back the diagonal block owned by this lane's column/half
            if (pb < pairs && (hi == (bblk >= 2))) {
                // rows 4*bblk..4*bblk+3 live in d[r] (bblk even-half) or d[4+r]
                const bool sel = (bblk & 1) != 0;
                float e0 = sel ? d[4] : d[0];
                float e1 = sel ? d[5] : d[1];
                float e2 = sel ? d[6] : d[2];
                float e3 = sel ? d[7] : d[3];
                mats[dst + pb][0 * 4 + bc] = e0;
                mats[dst + pb][1 * 4 + bc] = e1;
                mats[dst + pb][2 * 4 + bc] = e2;
                mats[dst + pb][3 * 4 + bc] = e3;
            }
        }
        __syncthreads();
        const int t = src; src = dst; dst = t;
        n = pairs;
    }

    // ---- close with radiation impedance; |Ze| = |A*ZL+B| / |C*ZL+D| ----
    if (lane == 0) {
        const float* fm = mats[src];
        const float Ar = fm[0],  Ai = fm[4];
        const float Br = fm[2],  Bi = fm[6];
        const float Cr = fm[8],  Ci = fm[12];
        const float Dr = fm[10], Di = fm[14];

        const float r_end = d1 * (1.0f / 2000.0f);
        const float Z0e   = RHO_AIR * C_SND / (PI_F * r_end * r_end);
        const float kr    = k * r_end;
        const float ZLr   = Z0e * 0.25f * kr * kr;
        const float ZLi   = Z0e * 0.61f * kr;

        float nr = Br, ni = Bi; cmad(nr, ni, Ar, Ai, ZLr, ZLi);
        float dr = Dr, di = Di; cmad(dr, di, Cr, Ci, ZLr, ZLi);

        out[fi] = sqrtf((nr * nr + ni * ni) / (dr * dr + di * di));
    }
}

extern "C" void kernel_launch(void* const* d_in, const int* in_sizes, int n_in,
                              void* d_out, int out_size, void* d_ws, size_t ws_size,
                              hipStream_t stream) {
    (void)in_sizes; (void)n_in; (void)d_ws; (void)ws_size;
    const float* plen  = (const float*)d_in[0];
    const float* pd1   = (const float*)d_in[1];
    const int*   pfmin = (const int*)  d_in[2];
    float* out = (float*)d_out;
    // one wave32 per frequency; F = out_size
    didge_impedance_kernel<<<dim3((unsigned)out_size), dim3(32), 0, stream>>>(
        plen, pd1, pfmin, out, out_size);
}